// BiaffineNER_46591805227174
// MI455X (gfx1250) — compile-verified
//
#include <hip/hip_runtime.h>

// Problem constants
#define BB   16
#define LC   512
#define LQ   64
#define HH   256
#define DD   1024          // 4*H
#define DFF  512
#define CC   3
#define DP   1025          // D+1
#define DPAD 1028          // D+1 padded to multiple of 4 (K granularity of wmma f32 16x16x4)
#define MROWS (BB*LC)      // 8192

typedef float v2f __attribute__((ext_vector_type(2)));
typedef float v8f __attribute__((ext_vector_type(8)));

// ---------------------------------------------------------------------------
// LDS-tiled WMMA f32 GEMM core.
// Block: 256 threads = 8 waves (4 M-waves x 2 N-waves). Block tile 128(M) x 64(N),
// K-step 16. Each wave owns a 32x32 sub-tile: 4 f32 16x16 accumulators,
// 2 A-fragments x 2 B-fragments per K=4 step -> 1 ds_load_b64 per WMMA.
// LDS tiles are k-contiguous per row (padded to 20 floats = 80B) so each
// fragment is one aligned float2 (single ds_load_b64, even VGPR pair, and the
// 32-lane b64 access covers all 64 LDS banks exactly once).
// Fragment layouts per CDNA5 ISA (CDNA5 ISA):
//   A 16x4 : lanes 0-15 -> M=lane, v0=K+0, v1=K+1 ; lanes 16-31 -> M=lane-16, v0=K+2, v1=K+3
//   B 4x16 : lanes 0-15 -> N=lane, v0=K+0, v1=K+1 ; lanes 16-31 -> N=lane-16, v0=K+2, v1=K+3
//   C/D    : VGPR r: lanes 0-15 -> (M=r, N=lane) ; lanes 16-31 -> (M=r+8, N=lane-16)
// ---------------------------------------------------------------------------
template <typename FA, typename FB, typename FE>
__device__ __forceinline__ void gemm_core(int K, FA loadA, FB loadB, FE epi) {
  __shared__ float As[128][20];  // [m][k], k-contiguous, +4 pad
  __shared__ float Bs[64][20];   // [n][k], k-contiguous, +4 pad

  const int tid  = threadIdx.x;
  const int lane = tid & 31;
  const int wave = tid >> 5;
  const int wm   = (wave & 3) << 5;   // wave row offset in block tile (0..96)
  const int wn   = (wave >> 2) << 5;  // wave col offset in block tile (0 or 32)
  const int l    = lane & 15;
  const int kh   = (lane >> 4) << 1;  // 0 for lanes 0-15, 2 for lanes 16-31

  v8f acc00 = {}, acc01 = {}, acc10 = {}, acc11 = {};

  for (int k0 = 0; k0 < K; k0 += 16) {
    __syncthreads();
#pragma unroll
    for (int r = 0; r < 8; ++r) {          // A tile: 128x16 = 2048 elements
      int idx = tid + (r << 8);
      As[idx >> 4][idx & 15] = loadA(idx >> 4, k0 + (idx & 15));
    }
#pragma unroll
    for (int r = 0; r < 4; ++r) {          // B tile: 16x64 = 1024, stored [n][k]
      int idx = tid + (r << 8);
      Bs[idx & 63][idx >> 6] = loadB(k0 + (idx >> 6), idx & 63);
    }
    __syncthreads();
#pragma unroll
    for (int kk = 0; kk < 16; kk += 4) {
      v2f a0 = *(const v2f*)&As[wm + l     ][kk + kh];
      v2f a1 = *(const v2f*)&As[wm + 16 + l][kk + kh];
      v2f b0 = *(const v2f*)&Bs[wn + l     ][kk + kh];
      v2f b1 = *(const v2f*)&Bs[wn + 16 + l][kk + kh];
      acc00 = __builtin_amdgcn_wmma_f32_16x16x4_f32(false, a0, false, b0,
                                                    (short)0, acc00, false, false);
      acc01 = __builtin_amdgcn_wmma_f32_16x16x4_f32(false, a0, false, b1,
                                                    (short)0, acc01, false, false);
      acc10 = __builtin_amdgcn_wmma_f32_16x16x4_f32(false, a1, false, b0,
                                                    (short)0, acc10, false, false);
      acc11 = __builtin_amdgcn_wmma_f32_16x16x4_f32(false, a1, false, b1,
                                                    (short)0, acc11, false, false);
    }
  }

  const int mo = (lane >> 4) << 3;  // +8 rows for upper half-wave
#pragma unroll
  for (int r = 0; r < 8; ++r) {
    epi(wm + r + mo,      wn + l,      acc00[r]);
    epi(wm + r + mo,      wn + 16 + l, acc01[r]);
    epi(wm + 16 + r + mo, wn + l,      acc10[r]);
    epi(wm + 16 + r + mo, wn + 16 + l, acc11[r]);
  }
}

// ---------------------------------------------------------------------------
// Attention-side kernels (VALU; tiny fraction of total FLOPs)
// ---------------------------------------------------------------------------
__global__ void qw2_kernel(const float* __restrict__ ques,
                           const float* __restrict__ w_sim,
                           float* __restrict__ qw2) {
  int idx = blockIdx.x * blockDim.x + threadIdx.x;  // b*LQ + j
  if (idx >= BB * LQ) return;
  const float* w2 = w_sim + HH;
  const float* q  = ques + (size_t)idx * HH;
  float s = 0.f;
#pragma unroll 8
  for (int h = 0; h < HH; ++h) s += q[h] * w2[h];
  qw2[idx] = s;
}

// One block (256 threads) per (b,i): sim row + softmax_j + c2q + rowmax.
// Writes x chunks 0..2 (ctx, c2q, ctx*c2q) directly.
__global__ void attn_row_kernel(const float* __restrict__ ctx,
                                const float* __restrict__ ques,
                                const float* __restrict__ w_sim,
                                const float* __restrict__ qw2,
                                float* __restrict__ xbuf,
                                float* __restrict__ rowmax) {
  const int row = blockIdx.x;          // b*LC + i
  const int b   = row >> 9;
  const int tid = threadIdx.x;

  __shared__ float cw3[HH];
  __shared__ float red[256];
  __shared__ float simj[LQ];
  __shared__ float m_s, sum_s;

  const float* crow = ctx + (size_t)row * HH;
  const float  cv   = crow[tid];
  cw3[tid] = cv * w_sim[2 * HH + tid];
  red[tid] = cv * w_sim[tid];
  __syncthreads();
  for (int s = 128; s > 0; s >>= 1) {
    if (tid < s) red[tid] += red[tid + s];
    __syncthreads();
  }
  const float cw1 = red[0];

  const float* qb = ques + (size_t)b * LQ * HH;
  if (tid < LQ) {
    const float* q = qb + (size_t)tid * HH;
    float s = 0.f;
#pragma unroll 8
    for (int h = 0; h < HH; ++h) s += cw3[h] * q[h];
    simj[tid] = s + cw1 + qw2[b * LQ + tid];
  }
  __syncthreads();

  if (tid < 32) {
    float m = fmaxf(simj[tid], simj[tid + 32]);
    for (int off = 16; off > 0; off >>= 1) m = fmaxf(m, __shfl_down(m, off, 32));
    if (tid == 0) m_s = m;
  }
  __syncthreads();
  const float mval = m_s;
  if (tid < LQ) simj[tid] = __expf(simj[tid] - mval);
  __syncthreads();
  if (tid < 32) {
    float s = simj[tid] + simj[tid + 32];
    for (int off = 16; off > 0; off >>= 1) s += __shfl_down(s, off, 32);
    if (tid == 0) sum_s = s;
  }
  __syncthreads();
  const float inv = 1.f / sum_s;

  // c2q[h], h = tid (coalesced reads of ques)
  float acc = 0.f;
#pragma unroll 8
  for (int j = 0; j < LQ; ++j) acc += simj[j] * qb[(size_t)j * HH + tid];
  acc *= inv;

  float* xr = xbuf + (size_t)row * DD;
  xr[tid]            = cv;
  xr[HH + tid]       = acc;
  xr[2 * HH + tid]   = cv * acc;
  if (tid == 0) rowmax[row] = mval;
}

// One block per b: softmax_i over rowmax, then q2c[b,h] = sum_i beta_i * ctx[b,i,h]
__global__ void q2c_kernel(const float* __restrict__ ctx,
                           const float* __restrict__ rowmax,
                           float* __restrict__ q2c) {
  const int b   = blockIdx.x;
  const int tid = threadIdx.x;  // 256
  __shared__ float beta[LC];
  __shared__ float red[256];

  const float* m = rowmax + b * LC;
  const float v0 = m[tid], v1 = m[tid + 256];
  red[tid] = fmaxf(v0, v1);
  __syncthreads();
  for (int s = 128; s > 0; s >>= 1) {
    if (tid < s) red[tid] = fmaxf(red[tid], red[tid + s]);
    __syncthreads();
  }
  const float M = red[0];
  __syncthreads();
  const float e0 = __expf(v0 - M), e1v = __expf(v1 - M);
  beta[tid] = e0; beta[tid + 256] = e1v;
  red[tid] = e0 + e1v;
  __syncthreads();
  for (int s = 128; s > 0; s >>= 1) {
    if (tid < s) red[tid] += red[tid + s];
    __syncthreads();
  }
  const float inv = 1.f / red[0];

  const float* cb = ctx + (size_t)b * LC * HH;
  float acc = 0.f;
  for (int i = 0; i < LC; ++i) acc += beta[i] * cb[(size_t)i * HH + tid];
  q2c[b * HH + tid] = acc * inv;
}

// Fill x chunk 3: ctx * q2c (broadcast over i)
__global__ void x4_kernel(const float* __restrict__ ctx,
                          const float* __restrict__ q2c,
                          float* __restrict__ xbuf) {
  int idx = blockIdx.x * 256 + threadIdx.x;  // over B*LC*H
  if (idx >= BB * LC * HH) return;
  int h   = idx & (HH - 1);
  int row = idx >> 8;
  int b   = row >> 9;
  xbuf[(size_t)row * DD + 3 * HH + h] = ctx[idx] * q2c[b * HH + h];
}

// s1 tail cols (1024->1, 1025..1027->0) and e1T tail rows (same) for padding
__global__ void tails_kernel(float* __restrict__ s1, float* __restrict__ e1T) {
  int idx = blockIdx.x * 256 + threadIdx.x;
  if (idx < MROWS * 4) {
    int r = idx >> 2, c = idx & 3;
    s1[(size_t)r * DPAD + DD + c] = (c == 0) ? 1.f : 0.f;
  }
  if (idx < BB * 4 * LC) {
    int b  = idx / (4 * LC);
    int rm = idx % (4 * LC);
    int jr = rm >> 9;         // 0..3 -> row 1024+jr
    int y  = rm & (LC - 1);
    e1T[((size_t)b * DPAD + DD + jr) * LC + y] = (jr == 0) ? 1.f : 0.f;
  }
}

// ---------------------------------------------------------------------------
// GEMM kernels built on gemm_core (block tile 128 x 64)
// ---------------------------------------------------------------------------
// h = relu(x @ W1 + b1)   [8192,1024]x[1024,512]
__global__ void ffw1_kernel(const float* __restrict__ x, const float* __restrict__ W1,
                            const float* __restrict__ b1, float* __restrict__ h) {
  const int m0 = blockIdx.y << 7, n0 = blockIdx.x << 6;
  gemm_core(DD,
      [&](int m, int k) { return x[(size_t)(m0 + m) * DD + k]; },
      [&](int k, int n) { return W1[(size_t)k * DFF + n0 + n]; },
      [&](int ml, int nl, float v) {
        float r = v + b1[n0 + nl];
        h[(size_t)(m0 + ml) * DFF + n0 + nl] = r > 0.f ? r : 0.f;
      });
}

// out = h @ W2 + b2   [8192,512]x[512,1024]
// TRANS=false: s1[m][n] (ld=DPAD).  TRANS=true: e1T[b][n][y] with m=b*512+y.
template <bool TRANS>
__global__ void ffw2_kernel(const float* __restrict__ h, const float* __restrict__ W2,
                            const float* __restrict__ b2, float* __restrict__ out) {
  const int m0 = blockIdx.y << 7, n0 = blockIdx.x << 6;
  gemm_core(DFF,
      [&](int m, int k) { return h[(size_t)(m0 + m) * DFF + k]; },
      [&](int k, int n) { return W2[(size_t)k * DD + n0 + n]; },
      [&](int ml, int nl, float v) {
        int m = m0 + ml, n = n0 + nl;
        float r = v + b2[n];
        if (!TRANS) {
          out[(size_t)m * DPAD + n] = r;
        } else {
          int b = m >> 9, y = m & (LC - 1);
          out[((size_t)b * DPAD + n) * LC + y] = r;
        }
      });
}

// t[c][x][j] = sum_i s1_b[x,i] * Wb[i,c,j]   (K=i padded to 1028; j tiles to 1088 w/ guards)
__global__ void biaf1_kernel(const float* __restrict__ s1b, const float* __restrict__ Wb,
                             float* __restrict__ t) {
  const int c  = blockIdx.z;
  const int m0 = blockIdx.y << 7, n0 = blockIdx.x << 6;
  gemm_core(DPAD,
      [&](int m, int k) { return (k < DPAD) ? s1b[(size_t)(m0 + m) * DPAD + k] : 0.f; },
      [&](int k, int n) {
        int nn = n0 + n;
        return (k < DP && nn < DP) ? Wb[((size_t)k * CC + c) * DP + nn] : 0.f;
      },
      [&](int ml, int nl, float v) {
        int n = n0 + nl;
        if (n < DPAD) t[((size_t)c * LC + m0 + ml) * DPAD + n] = v;
      });
}

// score[b,x,y,c] = sum_j t[c][x][j] * e1T_b[j][y]
__global__ void biaf2_kernel(const float* __restrict__ t, const float* __restrict__ e1Tb,
                             float* __restrict__ out, int b) {
  const int c  = blockIdx.z;
  const int m0 = blockIdx.y << 7, n0 = blockIdx.x << 6;
  const float* tc = t + (size_t)c * LC * DPAD;
  gemm_core(DPAD,
      [&](int m, int k) { return (k < DPAD) ? tc[(size_t)(m0 + m) * DPAD + k] : 0.f; },
      [&](int k, int n) { return (k < DPAD) ? e1Tb[(size_t)k * LC + n0 + n] : 0.f; },
      [&](int ml, int nl, float v) {
        out[(((size_t)b * LC + m0 + ml) * LC + (n0 + nl)) * CC + c] = v;
      });
}

// ---------------------------------------------------------------------------
extern "C" void kernel_launch(void* const* d_in, const int* in_sizes, int n_in,
                              void* d_out, int out_size, void* d_ws, size_t ws_size,
                              hipStream_t stream) {
  const float* ctx   = (const float*)d_in[0];
  const float* ques  = (const float*)d_in[1];
  const float* w_sim = (const float*)d_in[2];
  const float* W1s   = (const float*)d_in[3];
  const float* b1s   = (const float*)d_in[4];
  const float* W2s   = (const float*)d_in[5];
  const float* b2s   = (const float*)d_in[6];
  const float* W1e   = (const float*)d_in[7];
  const float* b1e   = (const float*)d_in[8];
  const float* W2e   = (const float*)d_in[9];
  const float* b2e   = (const float*)d_in[10];
  const float* Wb    = (const float*)d_in[11];
  float* out = (float*)d_out;

  float* ws   = (float*)d_ws;
  float* xbuf = ws;                          // 8,388,608
  float* hbuf = xbuf + (size_t)MROWS * DD;   // 4,194,304 (later reused as t)
  float* s1   = hbuf + (size_t)MROWS * DFF;  // 8,421,376
  float* e1T  = s1 + (size_t)MROWS * DPAD;   // 8,421,376
  float* qw2  = e1T + (size_t)BB * DPAD * LC;
  float* rowm = qw2 + BB * LQ;
  float* q2c  = rowm + MROWS;
  float* tbuf = hbuf;                        // alias: h dead once biaffine starts

  // ---- attention / feature assembly ----
  qw2_kernel<<<(BB * LQ + 255) / 256, 256, 0, stream>>>(ques, w_sim, qw2);
  attn_row_kernel<<<MROWS, 256, 0, stream>>>(ctx, ques, w_sim, qw2, xbuf, rowm);
  q2c_kernel<<<BB, 256, 0, stream>>>(ctx, rowm, q2c);
  x4_kernel<<<(BB * LC * HH) / 256, 256, 0, stream>>>(ctx, q2c, xbuf);
  tails_kernel<<<(MROWS * 4 + 255) / 256, 256, 0, stream>>>(s1, e1T);

  // ---- start / end feed-forward (WMMA f32 GEMMs) ----
  ffw1_kernel<<<dim3(DFF / 64, MROWS / 128), 256, 0, stream>>>(xbuf, W1s, b1s, hbuf);
  ffw2_kernel<false><<<dim3(DD / 64, MROWS / 128), 256, 0, stream>>>(hbuf, W2s, b2s, s1);
  ffw1_kernel<<<dim3(DFF / 64, MROWS / 128), 256, 0, stream>>>(xbuf, W1e, b1e, hbuf);
  ffw2_kernel<true><<<dim3(DD / 64, MROWS / 128), 256, 0, stream>>>(hbuf, W2e, b2e, e1T);

  // ---- biaffine, per batch (t buffer reused; stream-serialized) ----
  for (int b = 0; b < BB; ++b) {
    biaf1_kernel<<<dim3((DP + 63) / 64, LC / 128, CC), 256, 0, stream>>>(
        s1 + (size_t)b * LC * DPAD, Wb, tbuf);
    biaf2_kernel<<<dim3(LC / 64, LC / 128, CC), 256, 0, stream>>>(
        tbuf, e1T + (size_t)b * DPAD * LC, out, b);
  }
}